// LocationbasedAttention_9045201125788
// MI455X (gfx1250) — compile-verified
//
#include <hip/hip_runtime.h>
#include <hip/hip_bf16.h>

// Problem constants (fixed by the reference setup_inputs()).
#define BB 16
#define TT 2048
#define DD 512
#define NC 16           // chunks along T
#define TC (TT / NC)    // 128 rows per chunk

typedef __attribute__((ext_vector_type(2))) float v2f;
typedef __attribute__((ext_vector_type(8))) float v8f;

// ---------------------------------------------------------------------------
// K1: scores s[b,t] = x[b,t,:]·w + bias, then e = exp(s), via fp32 WMMA.
// Wave handles a 16-row tile: A(16x4 of x) x B(4x16, w broadcast to all cols).
// Every column of the 16x16 f32 accumulator holds s[m] after the K loop.
// ---------------------------------------------------------------------------
__global__ void __launch_bounds__(256)
k1_scores_wmma(const float* __restrict__ x, const float* __restrict__ w,
               const float* __restrict__ bias, float* __restrict__ e) {
    __shared__ float wl[DD];
    for (int i = threadIdx.x; i < DD; i += 256) wl[i] = w[i];
    __syncthreads();

    const int bt   = blockIdx.x;        // BB * (TT/128) = 256 blocks
    const int b    = bt >> 4;
    const int tile = bt & 15;
    const int wave = threadIdx.x >> 5;
    const int lane = threadIdx.x & 31;
    const int t0   = tile * 128 + wave * 16;

    const int m  = lane & 15;           // A: row index within tile
    const int kh = lane >> 4;           // A/B: K half (k offsets 2*kh, 2*kh+1)

    const float* rowp = x + ((size_t)b * TT + (size_t)(t0 + m)) * DD + 2 * kh;

    v8f c = {};
    for (int k = 0; k < DD; k += 4) {
        v2f a  = *(const v2f*)(rowp + k);              // A[m][k+2kh .. +1]
        v2f bw = *(const v2f*)(&wl[k + 2 * kh]);       // B[k'][n] = w[k'] (all n)
        c = __builtin_amdgcn_wmma_f32_16x16x4_f32(
                /*neg_a=*/false, a, /*neg_b=*/false, bw,
                /*c_mod=*/(short)0, c, /*reuse_a=*/false, /*reuse_b=*/false);
    }

    // C layout: lane l holds C[M = r + 8*(l>>4)][N = l&15], columns identical.
    const float bb = bias[0];
    if (m == 0) {
        float* eo = e + (size_t)b * TT + t0 + kh * 8;
#pragma unroll
        for (int r = 0; r < 8; ++r) eo[r] = __expf(c[r] + bb);
    }
}

// ---------------------------------------------------------------------------
// K2: per-batch inclusive cumsum of e -> store reciprocals invZ[b,t].
// One 256-thread block per batch; 8 elements per thread + LDS block scan.
// ---------------------------------------------------------------------------
__global__ void __launch_bounds__(256)
k2_zscan(const float* __restrict__ e, float* __restrict__ invZ) {
    __shared__ float part[256];
    const int b   = blockIdx.x;
    const int tid = threadIdx.x;
    const int base = tid * 8;
    const float* eb = e + (size_t)b * TT;

    float v[8];
    float run = 0.f;
#pragma unroll
    for (int i = 0; i < 8; ++i) { run += eb[base + i]; v[i] = run; }
    part[tid] = run;
    __syncthreads();
    if (tid == 0) {
        float acc = 0.f;
        for (int i = 0; i < 256; ++i) { float t = part[i]; part[i] = acc; acc += t; }
    }
    __syncthreads();
    const float off = part[tid];
    float* zb = invZ + (size_t)b * TT;
#pragma unroll
    for (int i = 0; i < 8; ++i) zb[base + i] = 1.0f / (off + v[i]);
}

// ---------------------------------------------------------------------------
// K3: per-chunk partial sums S[b,c,d] = sum_{t in chunk c} e[b,t]*x[b,t,d].
// grid (D/128, NC, B), 128 threads; coalesced streaming over x.
// ---------------------------------------------------------------------------
__global__ void __launch_bounds__(128)
k3_partials(const float* __restrict__ x, const float* __restrict__ e,
            float* __restrict__ S) {
    const int d = blockIdx.x * 128 + threadIdx.x;
    const int c = blockIdx.y;
    const int b = blockIdx.z;

    const float* xp = x + ((size_t)b * TT + (size_t)c * TC) * DD + d;
    const float* ep = e + (size_t)b * TT + (size_t)c * TC;

    float acc = 0.f;
    for (int tt = 0; tt < TC; ++tt) {
        __builtin_prefetch(&xp[(size_t)(tt + 8) * DD], 0, 1);  // global_prefetch_b8
        acc = fmaf(ep[tt], xp[(size_t)tt * DD], acc);
    }
    S[((size_t)(b * NC + c)) * DD + d] = acc;
}

// ---------------------------------------------------------------------------
// K4: in-place exclusive prefix over the NC chunks for each (b,d) column.
// ---------------------------------------------------------------------------
__global__ void __launch_bounds__(256)
k4_chunkprefix(float* __restrict__ S) {
    const int gid = blockIdx.x * 256 + threadIdx.x;   // BB*DD = 8192
    const int b = gid >> 9;
    const int d = gid & (DD - 1);
    float run = 0.f;
#pragma unroll
    for (int c = 0; c < NC; ++c) {
        const size_t idx = ((size_t)(b * NC + c)) * DD + d;
        const float t = S[idx];
        S[idx] = run;
        run += t;
    }
}

// ---------------------------------------------------------------------------
// K5: within-chunk weighted scan with chunk carry; out[b,j+1,d]=acc*invZ[b,j].
// Output is never re-read -> nontemporal stores keep x resident in L2.
// ---------------------------------------------------------------------------
__global__ void __launch_bounds__(128)
k5_scanout(const float* __restrict__ x, const float* __restrict__ e,
           const float* __restrict__ invZ, const float* __restrict__ S,
           float* __restrict__ out) {
    const int d = blockIdx.x * 128 + threadIdx.x;
    const int c = blockIdx.y;
    const int b = blockIdx.z;

    const size_t xbase = ((size_t)b * TT + (size_t)c * TC) * DD + d;
    const float* xp = x + xbase;
    const float* ep = e + (size_t)b * TT + (size_t)c * TC;
    const float* zp = invZ + (size_t)b * TT + (size_t)c * TC;

    float acc = S[((size_t)(b * NC + c)) * DD + d];
    float* op = out + xbase + DD;       // out[b, j+1, d], j = c*TC + tt

    if (c == 0) {  // t == 0 passthrough row
        __builtin_nontemporal_store(xp[0], out + (size_t)b * TT * DD + d);
    }
    const int jbase = c * TC;
    for (int tt = 0; tt < TC; ++tt) {
        __builtin_prefetch(&xp[(size_t)(tt + 8) * DD], 0, 1);
        const float xv = xp[(size_t)tt * DD];
        acc = fmaf(ep[tt], xv, acc);
        if (jbase + tt < TT - 1) {      // j = TT-1 has no output row
            __builtin_nontemporal_store(acc * zp[tt], op + (size_t)tt * DD);
        }
    }
}

// ---------------------------------------------------------------------------
extern "C" void kernel_launch(void* const* d_in, const int* in_sizes, int n_in,
                              void* d_out, int out_size, void* d_ws, size_t ws_size,
                              hipStream_t stream) {
    const float* x    = (const float*)d_in[0];   // [B,T,D]
    const float* w    = (const float*)d_in[1];   // [D,1]
    const float* bias = (const float*)d_in[2];   // [1]
    float* out = (float*)d_out;                  // [B,T,D]

    char* ws = (char*)d_ws;
    float* e    = (float*)(ws);                                   // B*T   (128 KB)
    float* invZ = (float*)(ws + (size_t)BB * TT * sizeof(float)); // B*T   (128 KB)
    float* S    = (float*)(ws + (size_t)2 * BB * TT * sizeof(float)); // B*NC*D (512 KB)

    k1_scores_wmma<<<BB * (TT / 128), 256, 0, stream>>>(x, w, bias, e);
    k2_zscan<<<BB, 256, 0, stream>>>(e, invZ);
    k3_partials<<<dim3(DD / 128, NC, BB), 128, 0, stream>>>(x, e, S);
    k4_chunkprefix<<<(BB * DD) / 256, 256, 0, stream>>>(S);
    k5_scanout<<<dim3(DD / 128, NC, BB), 128, 0, stream>>>(x, e, invZ, S, out);
}